// GNNLayer_25615184954165
// MI455X (gfx1250) — compile-verified
//
#include <hip/hip_runtime.h>

typedef __attribute__((ext_vector_type(2))) float v2f;
typedef __attribute__((ext_vector_type(8))) float v8f;

#define N_NODES 50000
#define N_EDGES 1600000
#define DIM     128
#define NREL    8
#define BCOLS   1280          // 8*128 relation cols + 128 loop + 128 res
#define BN_EPS  1e-5f
#define PANEL   128           // cols per block = whole output panel
#define LSTRIDE 132           // LDS col stride: (132*c + k) % 64 = (4c + k) % 64 -> conflict-free b64

// ---------------- K0: build combined B^T matrix (BCOLS x DIM) --------------
// Row c of B^T = output column c: [0,1024) basis-decomposed W_r ;
// [1024,1152) loop_w ; [1152,1280) res_w.  Bt[c][i] = B[i][c].
__global__ void build_Bt(const float* __restrict__ basis_w,   // (R, D, D)
                         const float* __restrict__ w_comp,    // (R, R)
                         const float* __restrict__ loop_w,    // (D, D)
                         const float* __restrict__ res_w,     // (D, D)
                         float* __restrict__ Bt) {
  int idx = blockIdx.x * blockDim.x + threadIdx.x;
  if (idx >= DIM * BCOLS) return;
  int c = idx / DIM;        // output column (row of B^T)
  int i = idx % DIM;        // K index (input channel)
  float v;
  if (c < NREL * DIM) {
    int r = c >> 7, o = c & 127;
    float acc = 0.f;
#pragma unroll
    for (int b = 0; b < NREL; ++b)
      acc = fmaf(w_comp[r * NREL + b], basis_w[(b * DIM + i) * DIM + o], acc);
    v = acc;
  } else if (c < NREL * DIM + DIM) {
    v = loop_w[i * DIM + (c - NREL * DIM)];
  } else {
    v = res_w[i * DIM + (c - NREL * DIM - DIM)];
  }
  Bt[idx] = v;
}

// ---------------- K1: fp32 WMMA GEMM with async-staged LDS B panel ---------
// Block = 128 threads = 4 waves. Block tile: 64 rows x 128 cols.
// Wave w owns rows [row0, row0+16) and ALL 128 cols (8 accumulators).
// B^T panel (contiguous 64 KB of global) staged once via
// GLOBAL_LOAD_ASYNC_TO_LDS_B128 (ASYNCcnt -> s_wait_asynccnt), stored
// column-major in LDS; each B fragment is then ONE conflict-free ds_load_b64.
__global__ __launch_bounds__(128) void gemm_wmma(
    const float* __restrict__ A,      // N x DIM
    const float* __restrict__ Bt,     // BCOLS x DIM (transposed B)
    float* __restrict__ C,            // N x DIM
    int colBase) {
  __shared__ float Blds[PANEL * LSTRIDE];   // 67.6 KB, [col][k] padded

  const int tid = threadIdx.x;

  // ---- async stage: 128x128 panel, each thread moves 32 float4 chunks ----
#pragma unroll 4
  for (int i = 0; i < 32; ++i) {
    int idx = tid + 128 * i;            // float4 index
    int c   = idx >> 5;                 // column (row of B^T)
    int k4  = (idx & 31) << 2;          // 4 consecutive K values
    const float* gsrc = Bt + (size_t)(colBase + c) * DIM + k4;
    // ptrtoint of the shared address escapes Blds so the memory-clobbering
    // asm is a legal writer; low 32 bits of the flat address = LDS byte offset.
    unsigned ldst = (unsigned)(size_t)(&Blds[c * LSTRIDE + k4]);
    asm volatile("global_load_async_to_lds_b128 %0, %1, off"
                 :: "v"(ldst), "v"(gsrc) : "memory");
  }
  asm volatile("s_wait_asynccnt 0x0" ::: "memory");
  __syncthreads();

  const int lane = tid & 31;
  const int wave = tid >> 5;            // 0..3 -> 16-row slice
  const int row0 = blockIdx.x * 64 + wave * 16;
  const int half = lane >> 4;           // frag half: 0 -> K=k,k+1 ; 1 -> K=k+2,k+3
  const int m    = lane & 15;

  int ar = row0 + m;
  ar = ar < N_NODES ? ar : N_NODES - 1;         // clamp tail rows (stores guarded)
  const float* arow = A + (size_t)ar * DIM;
  const float* bcol = &Blds[m * LSTRIDE];       // this lane's column base

  v8f acc[8];
#pragma unroll
  for (int t = 0; t < 8; ++t) acc[t] = (v8f){};

#pragma unroll 4
  for (int k = 0; k < DIM; k += 4) {
    const int ka = k + 2 * half;
    v2f a;
    a.x = arow[ka];
    a.y = arow[ka + 1];
#pragma unroll
    for (int t = 0; t < 8; ++t) {
      // {B[ka][col], B[ka+1][col]} contiguous in LDS -> single ds_load_b64
      v2f b = *(const v2f*)(bcol + t * 16 * LSTRIDE + ka);
      acc[t] = __builtin_amdgcn_wmma_f32_16x16x4_f32(false, a, false, b,
                                                     (short)0, acc[t], false, false);
    }
  }

  // C/D layout: VGPR j -> M = j + 8*half, N = lane&15
  if (blockIdx.x * 64 + 64 <= N_NODES) {          // block-uniform full tile
#pragma unroll
    for (int j = 0; j < 8; ++j) {
      float* crow = C + (size_t)(row0 + j + 8 * half) * DIM + m;
#pragma unroll
      for (int t = 0; t < 8; ++t) crow[t * 16] = acc[t][j];
    }
  } else {
#pragma unroll
    for (int j = 0; j < 8; ++j) {
      int row = row0 + j + 8 * half;
      if (row < N_NODES) {
        float* crow = C + (size_t)row * DIM + m;
#pragma unroll
        for (int t = 0; t < 8; ++t) crow[t * 16] = acc[t][j];
      }
    }
  }
}

// ---------------- K2: float4 fill ----------------
__global__ void fill_zero4(float4* __restrict__ p, int n4) {
  int i = blockIdx.x * blockDim.x + threadIdx.x;
  if (i < n4) p[i] = make_float4(0.f, 0.f, 0.f, 0.f);
}

// ---------------- K3: per-relation edge scatter, one wave per edge ---------
// 32 lanes x float4 -> one coalesced 512B row gather, wave-uniform early-out.
__global__ __launch_bounds__(256) void scatter_rel(
    const int* __restrict__ src, const int* __restrict__ dst,
    const int* __restrict__ etype,
    const float* __restrict__ xWr,   // N x DIM (relation r, L2-resident)
    float* __restrict__ agg,         // N x DIM
    int rel) {
  int tid  = blockIdx.x * 256 + threadIdx.x;
  int e    = tid >> 5;               // wave-uniform edge id
  int lane = tid & 31;
  if (e >= N_EDGES) return;
  if (etype[e] != rel) return;       // uniform scalar branch per wave
  const float4* srow = (const float4*)(xWr + (size_t)src[e] * DIM);
  float*        drow = agg + (size_t)dst[e] * DIM + lane * 4;
  float4 v = srow[lane];
  unsafeAtomicAdd(drow + 0, v.x);
  unsafeAtomicAdd(drow + 1, v.y);
  unsafeAtomicAdd(drow + 2, v.z);
  unsafeAtomicAdd(drow + 3, v.w);
}

// ---------------- K4: h = relu(agg + xloop + bias) + relu(xres + res_b) ----
__global__ void combine4(const float4* __restrict__ agg, const float4* __restrict__ xloop,
                         const float4* __restrict__ xres, const float4* __restrict__ bias,
                         const float4* __restrict__ res_b, float4* __restrict__ out) {
  int i = blockIdx.x * blockDim.x + threadIdx.x;   // float4 index, exact grid
  int d4 = i & 31;                                 // channel group (DIM/4 = 32)
  float4 g = agg[i], l = xloop[i], r = xres[i];
  float4 bb = bias[d4], rb = res_b[d4];
  float4 o;
  o.x = fmaxf(g.x + l.x + bb.x, 0.f) + fmaxf(r.x + rb.x, 0.f);
  o.y = fmaxf(g.y + l.y + bb.y, 0.f) + fmaxf(r.y + rb.y, 0.f);
  o.z = fmaxf(g.z + l.z + bb.z, 0.f) + fmaxf(r.z + rb.z, 0.f);
  o.w = fmaxf(g.w + l.w + bb.w, 0.f) + fmaxf(r.w + rb.w, 0.f);
  out[i] = o;
}

// ---------------- K5a: per-channel sum / sumsq (float4, stride % 32 == 0) ---
__global__ void bn_stats(const float* __restrict__ h, float* __restrict__ stats) {
  int t  = blockIdx.x * blockDim.x + threadIdx.x;
  int d4 = (t & 31) * 4;             // constant per thread: stride multiple of 32 float4
  const float4* h4 = (const float4*)h;
  float4 s  = make_float4(0.f, 0.f, 0.f, 0.f);
  float4 sq = make_float4(0.f, 0.f, 0.f, 0.f);
  size_t stride = (size_t)gridDim.x * blockDim.x;
  size_t total4 = (size_t)N_NODES * DIM / 4;
  for (size_t i = t; i < total4; i += stride) {
    float4 v = h4[i];
    s.x += v.x; s.y += v.y; s.z += v.z; s.w += v.w;
    sq.x += v.x * v.x; sq.y += v.y * v.y; sq.z += v.z * v.z; sq.w += v.w * v.w;
  }
  unsafeAtomicAdd(&stats[d4 + 0], s.x);
  unsafeAtomicAdd(&stats[d4 + 1], s.y);
  unsafeAtomicAdd(&stats[d4 + 2], s.z);
  unsafeAtomicAdd(&stats[d4 + 3], s.w);
  unsafeAtomicAdd(&stats[DIM + d4 + 0], sq.x);
  unsafeAtomicAdd(&stats[DIM + d4 + 1], sq.y);
  unsafeAtomicAdd(&stats[DIM + d4 + 2], sq.z);
  unsafeAtomicAdd(&stats[DIM + d4 + 3], sq.w);
}

// ---------------- K5b: normalize in place (float4) --------------------------
__global__ void bn_apply4(float4* __restrict__ h, const float* __restrict__ stats,
                          const float4* __restrict__ gamma, const float4* __restrict__ beta) {
  int i = blockIdx.x * blockDim.x + threadIdx.x;   // float4 index, exact grid
  int d4 = i & 31;
  const float invN = 1.0f / (float)N_NODES;
  const float4* mean4 = (const float4*)stats;
  const float4* sq4   = (const float4*)(stats + DIM);
  float4 mu = mean4[d4], qq = sq4[d4];
  float4 g = gamma[d4], bt = beta[d4];
  float4 v = h[i];
  float m0 = mu.x * invN, m1 = mu.y * invN, m2 = mu.z * invN, m3 = mu.w * invN;
  float i0 = rsqrtf(qq.x * invN - m0 * m0 + BN_EPS);
  float i1 = rsqrtf(qq.y * invN - m1 * m1 + BN_EPS);
  float i2 = rsqrtf(qq.z * invN - m2 * m2 + BN_EPS);
  float i3 = rsqrtf(qq.w * invN - m3 * m3 + BN_EPS);
  v.x = (v.x - m0) * i0 * g.x + bt.x;
  v.y = (v.y - m1) * i1 * g.y + bt.y;
  v.z = (v.z - m2) * i2 * g.z + bt.z;
  v.w = (v.w - m3) * i3 * g.w + bt.w;
  h[i] = v;
}

extern "C" void kernel_launch(void* const* d_in, const int* in_sizes, int n_in,
                              void* d_out, int out_size, void* d_ws, size_t ws_size,
                              hipStream_t stream) {
  const float* x       = (const float*)d_in[0];   // (N, D)
  const int*   src     = (const int*)d_in[1];     // (E,)
  const int*   dst     = (const int*)d_in[2];     // (E,)
  const int*   etype   = (const int*)d_in[3];     // (E,)
  const float* basis_w = (const float*)d_in[4];   // (R, D, D)
  const float* w_comp  = (const float*)d_in[5];   // (R, R)
  const float* loop_w  = (const float*)d_in[6];   // (D, D)
  const float* bias    = (const float*)d_in[7];   // (D,)
  const float* res_w   = (const float*)d_in[8];   // (D, D)
  const float* res_b   = (const float*)d_in[9];   // (D,)
  const float* gamma   = (const float*)d_in[10];  // (D,)
  const float* beta    = (const float*)d_in[11];  // (D,)
  float* out = (float*)d_out;

  // Workspace layout (floats): ~103 MB total
  float* ws    = (float*)d_ws;
  float* Bt    = ws;                          // 1280*128 = 163,840
  float* xWr   = Bt    + (size_t)DIM * BCOLS; // 6,400,000
  float* agg   = xWr   + (size_t)N_NODES * DIM;
  float* xloop = agg   + (size_t)N_NODES * DIM;
  float* xres  = xloop + (size_t)N_NODES * DIM;
  float* stats = xres  + (size_t)N_NODES * DIM;  // 256 floats

  const int NT = 256;
  const int ND4 = N_NODES * DIM / 4;          // 1,600,000 (exact multiple of 256)

  // K0: combined B^T matrix
  build_Bt<<<(DIM * BCOLS + NT - 1) / NT, NT, 0, stream>>>(basis_w, w_comp, loop_w, res_w, Bt);

  // K2: zero agg + stats
  fill_zero4<<<ND4 / NT, NT, 0, stream>>>((float4*)agg, ND4);
  fill_zero4<<<1, 64, 0, stream>>>((float4*)stats, 2 * DIM / 4);

  dim3 ggrid((N_NODES + 63) / 64);            // 782 blocks, 4 waves each
  dim3 sgrid((size_t)N_EDGES * 32 / NT);      // 200,000 blocks: 1 wave per edge

  // Per-relation: GEMM (25.6 MB result stays L2-resident) then edge scatter
  for (int r = 0; r < NREL; ++r) {
    gemm_wmma<<<ggrid, 128, 0, stream>>>(x, Bt, xWr, r * DIM);
    scatter_rel<<<sgrid, NT, 0, stream>>>(src, dst, etype, xWr, agg, r);
  }

  // Self-loop and residual GEMMs
  gemm_wmma<<<ggrid, 128, 0, stream>>>(x, Bt, xloop, NREL * DIM);
  gemm_wmma<<<ggrid, 128, 0, stream>>>(x, Bt, xres, NREL * DIM + DIM);

  // Fuse aggregation + self-loop + bias + residual ReLU
  combine4<<<ND4 / NT, NT, 0, stream>>>((const float4*)agg, (const float4*)xloop,
                                        (const float4*)xres, (const float4*)bias,
                                        (const float4*)res_b, (float4*)out);

  // BatchNorm (training-mode batch stats over node axis)
  bn_stats<<<256, NT, 0, stream>>>(out, stats);   // stride 65536 float4 (mult of 32)
  bn_apply4<<<ND4 / NT, NT, 0, stream>>>((float4*)out, stats,
                                         (const float4*)gamma, (const float4*)beta);
}